// ETM_50826642981279
// MI455X (gfx1250) — compile-verified
//
#include <hip/hip_runtime.h>
#include <hip/hip_bf16.h>
#include <math.h>

// ---------------------------------------------------------------------------
// ETM forward for MI455X (gfx1250, wave32, WMMA f32 16x16x4).
// Shapes: B=1024, S=512, V=50000, E=300, H=800, T=50.
// ---------------------------------------------------------------------------

#define B_  1024
#define S_  512
#define V_  50000
#define E_  300
#define H_  800
#define T_  50
#define TPAD 64   // T padded to 4 WMMA n-tiles

typedef float v2f __attribute__((ext_vector_type(2)));
typedef float v8f __attribute__((ext_vector_type(8)));

// D = A(16x4, f32) x B(4x16, f32) + C(16x16, f32)
static __device__ inline v8f wmma4(v2f a, v2f b, v8f c) {
    return __builtin_amdgcn_wmma_f32_16x16x4_f32(
        /*neg_a=*/false, a, /*neg_b=*/false, b,
        /*c_mod=*/(short)0, c, /*reuse_a=*/false, /*reuse_b=*/false);
}

// --------------------------- K1: x[b,e] from ids ----------------------------
// x[b] = (1/cnt) * sum_{s, ids!=1,2} rho[ids[b,s]];  cnt counts duplicates.
__global__ void etm_xbar(const int* __restrict__ ids,
                         const float* __restrict__ rho,
                         float* __restrict__ x) {
    __shared__ int sid[S_];
    const int b = blockIdx.x;
    for (int s = threadIdx.x; s < S_; s += blockDim.x) sid[s] = ids[b * S_ + s];
    __syncthreads();
    const int t = threadIdx.x;
    float acc = 0.f;
    int cnt = 0;
    for (int s = 0; s < S_; ++s) {
        const int id = sid[s];              // uniform across block -> scalar branch
        if (id == 1 || id == 2) continue;
        ++cnt;
        if (t < E_) acc += rho[(size_t)id * E_ + t];
    }
    if (t < E_) x[(size_t)b * E_ + t] = acc / (float)cnt;
}

// ------------------ K2a: logits L[v,t] = rho[v,:]·alpha[t,:] ----------------
// One wave per 16-row slab, all 4 n-tiles per wave (A loaded once, 4 wmmas of
// ILP).  Padded cols (>=T) load a clamped valid address; their results are
// never stored, so no zeroing/branching is needed in the k-loop.
__global__ void etm_logits(const float* __restrict__ rho,
                           const float* __restrict__ alpha,
                           float* __restrict__ L) {
    const int mt = blockIdx.x;           // 0..3124
    const int lane = threadIdx.x;        // wave32
    const int m  = lane & 15;
    const int hi = lane >> 4;            // 0 or 1
    const int n  = lane & 15;
    const float* arow = rho + (size_t)(mt * 16 + m) * E_;
    const float* b0p = alpha + (size_t)(n)            * E_;
    const float* b1p = alpha + (size_t)(16 + n)       * E_;
    const float* b2p = alpha + (size_t)(32 + n)       * E_;
    const int   c3   = (48 + n < T_) ? (48 + n) : (T_ - 1);   // clamp
    const float* b3p = alpha + (size_t)c3 * E_;
    v8f c0 = {}, c1 = {}, c2 = {}, c3v = {};
    for (int k = 0; k < E_; k += 4) {
        const int k0 = k + 2 * hi;
        v2f a  = *(const v2f*)(arow + k0);
        v2f b0 = *(const v2f*)(b0p + k0);
        v2f b1 = *(const v2f*)(b1p + k0);
        v2f b2 = *(const v2f*)(b2p + k0);
        v2f b3 = *(const v2f*)(b3p + k0);
        c0  = wmma4(a, b0, c0);
        c1  = wmma4(a, b1, c1);
        c2  = wmma4(a, b2, c2);
        c3v = wmma4(a, b3, c3v);
    }
    for (int r = 0; r < 8; ++r) {
        const int row = mt * 16 + r + 8 * hi;      // vocab index
        float* Lr = L + (size_t)row * T_;
        Lr[n]      = c0[r];
        Lr[16 + n] = c1[r];
        Lr[32 + n] = c2[r];
        if (48 + n < T_) Lr[48 + n] = c3v[r];
    }
}

// ----------- K2b: per-topic max + sum(exp) over V (fixed-order tree) --------
__global__ void etm_beta_reduce(const float* __restrict__ L,
                                float* __restrict__ mz /* [2*T_] */) {
    __shared__ float red[256];
    const int k = blockIdx.x;            // topic
    float mx = -1e30f;
    for (int v = threadIdx.x; v < V_; v += 256)
        mx = fmaxf(mx, L[(size_t)v * T_ + k]);
    red[threadIdx.x] = mx; __syncthreads();
    for (int s = 128; s > 0; s >>= 1) {
        if (threadIdx.x < s) red[threadIdx.x] = fmaxf(red[threadIdx.x], red[threadIdx.x + s]);
        __syncthreads();
    }
    mx = red[0]; __syncthreads();
    float sum = 0.f;
    for (int v = threadIdx.x; v < V_; v += 256)
        sum += expf(L[(size_t)v * T_ + k] - mx);
    red[threadIdx.x] = sum; __syncthreads();
    for (int s = 128; s > 0; s >>= 1) {
        if (threadIdx.x < s) red[threadIdx.x] += red[threadIdx.x + s];
        __syncthreads();
    }
    if (threadIdx.x == 0) { mz[k] = mx; mz[T_ + k] = red[0]; }
}

// --------------- K2c: normalize logits in place -> beta^T [V,T] -------------
__global__ void etm_beta_norm(float* __restrict__ L,
                              const float* __restrict__ mz) {
    const int i = blockIdx.x * 256 + threadIdx.x;    // i = v*T_ + k
    if (i >= V_ * T_) return;
    const int k = i % T_;
    L[i] = expf(L[i] - mz[k]) / mz[T_ + k];
}

// -------------------- K3: h = relu(x @ W1 + b1)  (WMMA) ---------------------
// grid = (B/16, H/32): each wave computes two adjacent 16x16 n-tiles,
// sharing the A operand.
__global__ void etm_gemm_h(const float* __restrict__ x,
                           const float* __restrict__ W1,
                           const float* __restrict__ b1,
                           float* __restrict__ h) {
    const int mt = blockIdx.x;           // 0..63
    const int nt = blockIdx.y;           // 0..24  (pairs of n-tiles)
    const int lane = threadIdx.x;
    const int m  = lane & 15;
    const int hi = lane >> 4;
    const int n  = lane & 15;
    const int colA = nt * 32 + n;
    const int colB = colA + 16;
    const float* arow = x + (size_t)(mt * 16 + m) * E_;
    v8f ca = {}, cb = {};
    for (int k = 0; k < E_; k += 4) {
        const int k0 = k + 2 * hi;
        v2f a = *(const v2f*)(arow + k0);
        v2f ba, bb;
        ba.x = W1[(size_t)k0 * H_ + colA];
        ba.y = W1[(size_t)(k0 + 1) * H_ + colA];
        bb.x = W1[(size_t)k0 * H_ + colB];
        bb.y = W1[(size_t)(k0 + 1) * H_ + colB];
        ca = wmma4(a, ba, ca);
        cb = wmma4(a, bb, cb);
    }
    const float biasA = b1[colA];
    const float biasB = b1[colB];
    for (int r = 0; r < 8; ++r) {
        const int row = mt * 16 + r + 8 * hi;
        const float va = ca[r] + biasA;
        const float vb = cb[r] + biasB;
        h[(size_t)row * H_ + colA] = va > 0.f ? va : 0.f;
        h[(size_t)row * H_ + colB] = vb > 0.f ? vb : 0.f;
    }
}

// ------- K4: mu/lv tiles (WMMA), then per-row KL + softmax -> theta ---------
// grid = B/16, block = 128 (4 waves, one n-tile each; 64 >= T=50).  Padded
// cols use a clamped address; their LDS slots hold garbage that is never read.
__global__ void etm_head(const float* __restrict__ h,
                         const float* __restrict__ Wmu, const float* __restrict__ bmu,
                         const float* __restrict__ Wlv, const float* __restrict__ blv,
                         float* __restrict__ theta,     // d_out, [B,T]
                         float* __restrict__ kl) {      // [B]
    __shared__ float mu_s[16][TPAD];
    __shared__ float lv_s[16][TPAD];
    const int mt = blockIdx.x;
    const int w    = threadIdx.x >> 5;   // n-tile 0..3
    const int lane = threadIdx.x & 31;
    const int m  = lane & 15;
    const int hi = lane >> 4;
    const int n  = lane & 15;
    const int col  = w * 16 + n;         // 0..63
    const bool cv  = (col < T_);
    const int ccol = cv ? col : (T_ - 1);        // clamped load address
    const float* arow = h + (size_t)(mt * 16 + m) * H_;
    v8f cmu = {}, clv = {};
    for (int k = 0; k < H_; k += 4) {
        const int k0 = k + 2 * hi;
        v2f a = *(const v2f*)(arow + k0);
        v2f bm, bl;
        bm.x = Wmu[(size_t)k0 * T_ + ccol];
        bm.y = Wmu[(size_t)(k0 + 1) * T_ + ccol];
        bl.x = Wlv[(size_t)k0 * T_ + ccol];
        bl.y = Wlv[(size_t)(k0 + 1) * T_ + ccol];
        cmu = wmma4(a, bm, cmu);
        clv = wmma4(a, bl, clv);
    }
    const float bmu_c = bmu[ccol];
    const float blv_c = blv[ccol];
    for (int r = 0; r < 8; ++r) {
        mu_s[r + 8 * hi][col] = cmu[r] + bmu_c;
        lv_s[r + 8 * hi][col] = clv[r] + blv_c;
    }
    __syncthreads();
    if (threadIdx.x < 16) {
        const int row = mt * 16 + threadIdx.x;
        float mx = -1e30f, klacc = 0.f;
        for (int t = 0; t < T_; ++t) {
            const float mu = mu_s[threadIdx.x][t];
            const float lv = lv_s[threadIdx.x][t];
            klacc += 1.f + lv - mu * mu - expf(lv);
            mx = fmaxf(mx, mu);
        }
        kl[row] = -0.5f * klacc;
        float ssum = 0.f;
        for (int t = 0; t < T_; ++t) ssum += expf(mu_s[threadIdx.x][t] - mx);
        const float inv = 1.f / ssum;
        for (int t = 0; t < T_; ++t)
            theta[(size_t)row * T_ + t] = expf(mu_s[threadIdx.x][t] - mx) * inv;
    }
}

// ---- K5: recon[b] = -sum_s log(theta_b · beta[:,id] + 1e-5) over tokens ----
__global__ void etm_recon(const int* __restrict__ ids,
                          const float* __restrict__ betaT,   // [V,T]
                          const float* __restrict__ theta,   // [B,T]
                          float* __restrict__ recon) {
    __shared__ float th[T_];
    __shared__ float red[256];
    const int b = blockIdx.x;
    if (threadIdx.x < T_) th[threadIdx.x] = theta[(size_t)b * T_ + threadIdx.x];
    __syncthreads();
    float acc = 0.f;
    for (int s = threadIdx.x; s < S_; s += 256) {
        const int id = ids[b * S_ + s];
        if (id == 1 || id == 2) continue;
        const float* bc = betaT + (size_t)id * T_;
        float v = 0.f;
#pragma unroll
        for (int t = 0; t < T_; ++t) v += th[t] * bc[t];
        acc += logf(v + 1e-5f);
    }
    red[threadIdx.x] = acc; __syncthreads();
    for (int s2 = 128; s2 > 0; s2 >>= 1) {
        if (threadIdx.x < s2) red[threadIdx.x] += red[threadIdx.x + s2];
        __syncthreads();
    }
    if (threadIdx.x == 0) recon[b] = -red[0];
}

// ---------------- K6: loss = mean(recon) + mean(kl) -------------------------
__global__ void etm_loss(const float* __restrict__ kl,
                         const float* __restrict__ recon,
                         float* __restrict__ out_loss) {
    __shared__ float red[256];
    float s = 0.f;
    for (int i = threadIdx.x; i < B_; i += 256) s += kl[i] + recon[i];
    red[threadIdx.x] = s; __syncthreads();
    for (int t = 128; t > 0; t >>= 1) {
        if (threadIdx.x < t) red[threadIdx.x] += red[threadIdx.x + t];
        __syncthreads();
    }
    if (threadIdx.x == 0) out_loss[0] = red[0] / (float)B_;
}

// ---------------------------------------------------------------------------
extern "C" void kernel_launch(void* const* d_in, const int* in_sizes, int n_in,
                              void* d_out, int out_size, void* d_ws, size_t ws_size,
                              hipStream_t stream) {
    const int*   ids   = (const int*)  d_in[0];
    const float* rho   = (const float*)d_in[1];
    const float* alpha = (const float*)d_in[2];
    const float* W1    = (const float*)d_in[3];
    const float* b1    = (const float*)d_in[4];
    const float* Wmu   = (const float*)d_in[5];
    const float* bmu   = (const float*)d_in[6];
    const float* Wlv   = (const float*)d_in[7];
    const float* blv   = (const float*)d_in[8];

    float* out   = (float*)d_out;          // [B*T] theta, then [1] loss
    float* theta = out;
    float* loss  = out + (size_t)B_ * T_;

    // Workspace layout (floats)
    float* ws    = (float*)d_ws;
    float* x     = ws;                                   // B*E      = 307200
    float* h     = x  + (size_t)B_ * E_;                 // B*H      = 819200
    float* L     = h  + (size_t)B_ * H_;                 // V*T      = 2500000
    float* mz    = L  + (size_t)V_ * T_;                 // 2*T      = 100
    float* kl    = mz + 2 * T_;                          // B        = 1024
    float* recon = kl + B_;                              // B        = 1024

    // 1) bag-of-words embedding (sparse: gather rho rows, skip tokens 1,2)
    etm_xbar<<<B_, 320, 0, stream>>>(ids, rho, x);

    // 2) topic-word logits via WMMA, then softmax over V per topic
    etm_logits<<<V_ / 16, 32, 0, stream>>>(rho, alpha, L);
    etm_beta_reduce<<<T_, 256, 0, stream>>>(L, mz);
    etm_beta_norm<<<(V_ * T_ + 255) / 256, 256, 0, stream>>>(L, mz);

    // 3) encoder hidden layer via WMMA
    etm_gemm_h<<<dim3(B_ / 16, H_ / 32), 32, 0, stream>>>(x, W1, b1, h);

    // 4) mu / logsigma heads via WMMA + per-row KL and softmax(theta)
    etm_head<<<B_ / 16, 128, 0, stream>>>(h, Wmu, bmu, Wlv, blv, theta, kl);

    // 5) sparse reconstruction loss per doc (only tokens present in doc)
    etm_recon<<<B_, 256, 0, stream>>>(ids, L, theta, recon);

    // 6) final scalar loss
    etm_loss<<<1, 256, 0, stream>>>(kl, recon, loss);
}